// NodeFeatureTile_42717744726390
// MI455X (gfx1250) — compile-verified
//
#include <hip/hip_runtime.h>

typedef float v2f __attribute__((ext_vector_type(2)));
typedef float v8f __attribute__((ext_vector_type(8)));

#define BDIM 4
#define GDIM 16
#define NPTS 20000
#define CCH  32
#define TILE_PTS 16
#define NT   (NPTS / TILE_PTS)   // 1250 point-tiles per (b,g)

__device__ __forceinline__ float lane_bcast(float x, int lane) {
    return __uint_as_float(__builtin_amdgcn_readlane(__float_as_uint(x), (unsigned)lane));
}

// ---------------------------------------------------------------------------
// Transpose one level: [G, C=32, R*R] -> [G, R*R, C=32]  (LDS 32x33 tiles)
// ---------------------------------------------------------------------------
__global__ __launch_bounds__(256)
void NodeFeat_transpose_kernel(const float* __restrict__ in,
                               float* __restrict__ out, int R)
{
    __shared__ float tile[32][33];
    const int P = R * R;
    const int tilesPerG = P >> 5;
    const int bid = blockIdx.x;
    const int g  = bid / tilesPerG;
    const int p0 = (bid - g * tilesPerG) << 5;
    const int tx = threadIdx.x, ty = threadIdx.y;   // blockDim = (32, 8)

    const float* ip = in  + (size_t)g * CCH * P;
    float*       op = out + (size_t)g * P * CCH;

#pragma unroll
    for (int k = 0; k < 4; ++k) {
        const int c = ty + k * 8;
        tile[c][tx] = ip[(size_t)c * P + p0 + tx];   // coalesced over pixels
    }
    __syncthreads();
#pragma unroll
    for (int k = 0; k < 4; ++k) {
        const int r = ty + k * 8;
        op[(size_t)(p0 + r) * CCH + tx] = tile[tx][r];   // coalesced over channels
    }
}

// ---------------------------------------------------------------------------
// Fused projection (WMMA f32 16x16x4) + 4-level bilinear gather.
// One wave = one 16-point tile; lane = channel (wave32 == C=32).
// TRANS=true : featmaps are [G,R,R,C] (coalesced 128B taps)
// TRANS=false: featmaps are [G,C,R,R] (fallback, strided taps)
// ---------------------------------------------------------------------------
template <bool TRANS>
__global__ __launch_bounds__(256)
void NodeFeat_sample_kernel(const float* __restrict__ pts,
                            const float* __restrict__ proj,
                            const float* __restrict__ f0,
                            const float* __restrict__ f1,
                            const float* __restrict__ f2,
                            const float* __restrict__ f3,
                            float* __restrict__ out)
{
    const int L = threadIdx.x;                         // lane == channel
    const int tile = blockIdx.x * blockDim.y + threadIdx.y;  // wave-uniform
    const int totalTiles = BDIM * GDIM * NT;
    if (tile >= totalTiles) return;                    // wave-uniform branch

    const int n0 = (tile % NT) * TILE_PTS;
    const int bg = tile / NT;                          // b*G + g
    const int g  = bg % GDIM;
    const size_t ptBase = (size_t)bg * NPTS + n0;      // flat point index base

    // ---- A operand: 16x4 f32. A[m,k] = points[m][k] (k<3), A[m,3]=0.
    //      Layout: lanes 0-15 M=lane, lanes 16-31 M=lane-16;
    //      VGPR0 = K0|K2, VGPR1 = K1|K3.
    const int m  = L & 15;
    const size_t pb = (ptBase + m) * 3;
    const float ax = pts[pb + ((L < 16) ? 0 : 2)];
    const float ay = (L < 16) ? pts[pb + 1] : 0.0f;

    // ---- B operand: 4x16 f32. B[k,n] = proj[n][k] (n<2, k<3), else 0.
    //      VGPR0 = K0|K2, VGPR1 = K1|K3; lane%16 = N.
    const size_t q = (size_t)bg * 6;
    const float p00 = proj[q + 0], p01 = proj[q + 1], p02 = proj[q + 2];
    const float p10 = proj[q + 3], p11 = proj[q + 4], p12 = proj[q + 5];
    const int n = L & 15;
    const float bx = (L < 16) ? ((n == 0) ? p00 : (n == 1) ? p10 : 0.0f)
                              : ((n == 0) ? p02 : (n == 1) ? p12 : 0.0f);
    const float by = (L < 16) ? ((n == 0) ? p01 : (n == 1) ? p11 : 0.0f) : 0.0f;

    const v2f A = {ax, ay};
    const v2f B = {bx, by};
    v8f C = {0.f, 0.f, 0.f, 0.f, 0.f, 0.f, 0.f, 0.f};
    // D[m,n] = sum_k points[m,k] * proj[n,k]  (full fp32, EXEC all-1s here)
    v8f D = __builtin_amdgcn_wmma_f32_16x16x4_f32(
        false, A, false, B, (short)0, C, false, false);

    const float* fms[4] = {f0, f1, f2, f3};
    const int    Rs[4]  = {128, 64, 32, 16};

#pragma unroll
    for (int p = 0; p < 16; ++p) {
        // D layout: VGPR r holds M=r (lanes 0-15, N=lane) and M=r+8 (lanes 16-31)
        const float u  = lane_bcast(D[p & 7], (p < 8) ? 0 : 16);
        const float vv = lane_bcast(D[p & 7], (p < 8) ? 1 : 17);

        float acc = 0.0f;
#pragma unroll
        for (int lvl = 0; lvl < 4; ++lvl) {
            const int R = Rs[lvl];
            const float* __restrict__ fm = fms[lvl];
            const float s = 0.5f * (float)(R - 1);
            const float ix = fminf(fmaxf((u  + 1.0f) * s, 0.0f), (float)(R - 1));
            const float iy = fminf(fmaxf((vv + 1.0f) * s, 0.0f), (float)(R - 1));
            const float x0f = floorf(ix), y0f = floorf(iy);
            const int x0 = (int)x0f, y0 = (int)y0f;
            const int x1 = min(x0 + 1, R - 1), y1 = min(y0 + 1, R - 1);
            const float wx = ix - x0f, wy = iy - y0f;

            float v00, v01, v10, v11;
            if (TRANS) {
                const size_t gb = (size_t)g * R;
                const size_t r0 = (gb + y0) * R;
                const size_t r1 = (gb + y1) * R;
                v00 = fm[(r0 + x0) * CCH + L];   // one 128B cacheline per tap
                v01 = fm[(r0 + x1) * CCH + L];
                v10 = fm[(r1 + x0) * CCH + L];
                v11 = fm[(r1 + x1) * CCH + L];
            } else {
                const size_t cb = ((size_t)g * CCH + L) * R;
                v00 = fm[(cb + y0) * R + x0];
                v01 = fm[(cb + y0) * R + x1];
                v10 = fm[(cb + y1) * R + x0];
                v11 = fm[(cb + y1) * R + x1];
            }
            const float top = v00 + wx * (v01 - v00);
            const float bot = v10 + wx * (v11 - v10);
            acc += top + wy * (bot - top);
        }
        // streaming store: don't pollute L2 (featmaps must stay resident)
        __builtin_nontemporal_store(acc, &out[(ptBase + p) * CCH + L]);
    }
}

// ---------------------------------------------------------------------------
extern "C" void kernel_launch(void* const* d_in, const int* in_sizes, int n_in,
                              void* d_out, int out_size, void* d_ws, size_t ws_size,
                              hipStream_t stream)
{
    const float* pts  = (const float*)d_in[0];
    const float* proj = (const float*)d_in[1];
    const float* fm[4] = {(const float*)d_in[2], (const float*)d_in[3],
                          (const float*)d_in[4], (const float*)d_in[5]};
    float* out = (float*)d_out;

    const size_t lvlFloats[4] = {
        (size_t)GDIM * 128 * 128 * CCH,
        (size_t)GDIM * 64  * 64  * CCH,
        (size_t)GDIM * 32  * 32  * CCH,
        (size_t)GDIM * 16  * 16  * CCH};
    const size_t totalFloats =
        lvlFloats[0] + lvlFloats[1] + lvlFloats[2] + lvlFloats[3];

    const dim3 blk(32, 8);                       // 8 waves per block
    const int totalTiles = BDIM * GDIM * NT;     // 80,000
    const int nBlocks = (totalTiles + 7) / 8;

    if (ws_size >= totalFloats * sizeof(float)) {
        float* wsf = (float*)d_ws;
        float* ft[4];
        size_t off = 0;
        const int Rs[4] = {128, 64, 32, 16};
        for (int l = 0; l < 4; ++l) { ft[l] = wsf + off; off += lvlFloats[l]; }
        for (int l = 0; l < 4; ++l) {
            const int R = Rs[l];
            const int blocks = GDIM * (R * R / 32);
            NodeFeat_transpose_kernel<<<blocks, blk, 0, stream>>>(fm[l], ft[l], R);
        }
        NodeFeat_sample_kernel<true><<<nBlocks, blk, 0, stream>>>(
            pts, proj, ft[0], ft[1], ft[2], ft[3], out);
    } else {
        NodeFeat_sample_kernel<false><<<nBlocks, blk, 0, stream>>>(
            pts, proj, fm[0], fm[1], fm[2], fm[3], out);
    }
}